// OffsetGaussianFalloffKernel_14688788152368
// MI455X (gfx1250) — compile-verified
//
#include <hip/hip_runtime.h>
#include <hip/hip_bf16.h>

typedef float v2f __attribute__((ext_vector_type(2)));
typedef float v8f __attribute__((ext_vector_type(8)));

#define LOG2E 1.4426950408889634f

// Each wave32 produces velocities for a tile of 16 points.
// Strength S[p,s] is computed per-lane (2 trans + ~10 VALU per pair), and the
// contraction over the 32 vortices runs on the matrix pipe:
//   D = S(16x32) x B(32x16), B columns = (1, x_s, y_s, 0...)
// via 8 chained V_WMMA_F32_16X16X4_F32 (full f32 accuracy).
__global__ __launch_bounds__(256) void OffsetGaussianFalloffKernel_14688788152368_kernel(
    const float* __restrict__ vf,   // (32, 8): y,x,tau,sig,v,u,off,sig_l
    const float* __restrict__ pts,  // (npts, 2): y,x
    float* __restrict__ out,        // (npts, 2)
    int npts) {
  __shared__ float4 vA[32];       // y, x, tau, off
  __shared__ float2 vC[32];       // -log2e/sig_r^2, -log2e/sig_l^2
  __shared__ float  sOut[8][48];  // per-wave staging: 16 points x 3 channels

  const int tid = threadIdx.x;
  if (tid < 32) {
    const float* f = vf + tid * 8;
    float y = f[0], x = f[1], tau = f[2], sig = f[3], off = f[6], sig_l = f[7];
    float sr = sig - sig_l;
    vA[tid] = make_float4(y, x, tau, off);
    vC[tid] = make_float2(-LOG2E / (sr * sr), -LOG2E / (sig_l * sig_l));
  }
  __syncthreads();

  const int wave  = tid >> 5;
  const int lane  = tid & 31;
  const int col   = lane & 15;          // point-in-tile (lanes 0-15) / B column
  const int khalf = (lane >> 4) << 1;   // 0 or 2: K-offset per A-matrix layout

  const int p0 = (blockIdx.x * 8 + wave) * 16;
  const int pt = p0 + col;
  const int ptc = pt < npts ? pt : npts - 1;  // clamp so EXEC stays all-1s
  const float2 P = ((const float2*)pts)[ptc];
  const float py = P.x, px = P.y;

  // B-column one-hot selectors (precomputed: B built with 2 FMAs/iteration)
  const float m0 = (col == 0) ? 1.0f : 0.0f;
  const float m1 = (col == 1) ? 1.0f : 0.0f;
  const float m2 = (col == 2) ? 1.0f : 0.0f;

  v8f acc = {0.f, 0.f, 0.f, 0.f, 0.f, 0.f, 0.f, 0.f};

#pragma unroll
  for (int j = 0; j < 8; ++j) {
    const int k0 = 4 * j + khalf;
    const float4 qa = vA[k0];
    const float4 qb = vA[k0 + 1];
    const float2 ca = vC[k0];
    const float2 cb = vC[k0 + 1];

    // strength(point, k0): tau * exp(-(|d|-off)^2/sigma^2) / |d|
    float d0 = py - qa.x, d1 = px - qa.y;
    float sq = __builtin_fmaf(d0, d0, d1 * d1);
    float r  = __builtin_amdgcn_rsqf(sq);
    float od = sq * r - qa.w;                  // dist - offset
    float c  = od >= 0.0f ? ca.x : ca.y;       // -log2e/sigma^2 (side select)
    float e  = __builtin_amdgcn_exp2f(od * od * c);
    const float sa = qa.z * e * r;

    // strength(point, k0+1)
    d0 = py - qb.x; d1 = px - qb.y;
    sq = __builtin_fmaf(d0, d0, d1 * d1);
    r  = __builtin_amdgcn_rsqf(sq);
    od = sq * r - qb.w;
    c  = od >= 0.0f ? cb.x : cb.y;
    e  = __builtin_amdgcn_exp2f(od * od * c);
    const float sb = qb.z * e * r;

    v2f a; a.x = sa; a.y = sb;
    v2f b;
    b.x = __builtin_fmaf(m1, qa.y, __builtin_fmaf(m2, qa.x, m0));  // (1, x_s, y_s)
    b.y = __builtin_fmaf(m1, qb.y, __builtin_fmaf(m2, qb.x, m0));

    // D = A(16x4) x B(4x16) + C, chained accumulation over K=32
    acc = __builtin_amdgcn_wmma_f32_16x16x4_f32(false, a, false, b, (short)0,
                                                acc, false, false);
  }

  // Stage D channels 0..2 to LDS (D layout: VGPR r = row r (lanes 0-15) /
  // row r+8 (lanes 16-31), column = lane%16).
  const int mb = (lane >> 4) * 8;
  if (col < 3) {
#pragma unroll
    for (int r8 = 0; r8 < 8; ++r8)
      sOut[wave][(mb + r8) * 3 + col] = acc[r8];
  }
  __syncthreads();

  if (lane < 16 && pt < npts) {
    const float S0 = sOut[wave][lane * 3 + 0];  // sum strength
    const float Sx = sOut[wave][lane * 3 + 1];  // sum strength*x_s
    const float Sy = sOut[wave][lane * 3 + 2];  // sum strength*y_s
    const float vy = __builtin_fmaf(px, S0, -Sx);   //  sum S*(px-xs)
    const float vx = __builtin_fmaf(-py, S0, Sy);   // -sum S*(py-ys)
    ((float2*)out)[pt] = make_float2(vy, vx);
  }
}

extern "C" void kernel_launch(void* const* d_in, const int* in_sizes, int n_in,
                              void* d_out, int out_size, void* d_ws, size_t ws_size,
                              hipStream_t stream) {
  const float* vf  = (const float*)d_in[0];  // vortex_feature (1,32,8) f32
  const float* pts = (const float*)d_in[1];  // points (1,1024,1024,2) f32
  float* out = (float*)d_out;                // (1,1024,1024,2) f32

  const int npts = in_sizes[1] / 2;          // 1,048,576
  const int blocks = (npts + 127) / 128;     // 128 points per 256-thread block
  OffsetGaussianFalloffKernel_14688788152368_kernel<<<blocks, 256, 0, stream>>>(
      vf, pts, out, npts);
}